// DualBranchCorrectionNet_62989990363311
// MI455X (gfx1250) — compile-verified
//
#include <hip/hip_runtime.h>

typedef __attribute__((ext_vector_type(2))) float v2f;
typedef __attribute__((ext_vector_type(8))) float v8f;

#define BATCH  16
#define NATOMS 50000
#define NBONDS 100000
#define COLS   (NATOMS * 3)   // 150000
#define KDIM   256
#define HID    128

// ---------------- workspace layout (float offsets) ----------------
#define WS_X    0u          // 16*256 = 4096        (trunk activations x)
#define WS_C    4096u       // 12  (folded 3x3 C + cvec[3])
#define WS_DEG  4112u       // 50000                 (degree, float)
#define WS_SUM  54112u      // 2,400,000             (neighbor scatter sums)
#define WS_H    2454112u    // 2,400,000             (graph hidden state h)
// total ~4.86M floats = ~19.4 MB

__global__ void zero_kernel(float* __restrict__ p, int n) {
  int i = blockIdx.x * blockDim.x + threadIdx.x;
  int stride = gridDim.x * blockDim.x;
  for (; i < n; i += stride) p[i] = 0.0f;
}

// Collapse msg (3->128) and upd (128->3) into C[3][3] + cvec[3].
// C[i][j] = sum_k msg_w[k][i] * upd_w[j][k];  cvec[j] = sum_k msg_b[k] * upd_w[j][k]
__global__ void fold_kernel(const float* __restrict__ msg_w,
                            const float* __restrict__ msg_b,
                            const float* __restrict__ upd_w,
                            float* __restrict__ Cbuf) {
  int t = threadIdx.x;
  if (t < 9) {
    int i = t / 3, j = t % 3;
    float s = 0.0f;
    for (int k = 0; k < HID; ++k) s += msg_w[k * 3 + i] * upd_w[j * HID + k];
    Cbuf[i * 3 + j] = s;
  } else if (t < 12) {
    int j = t - 9;
    float s = 0.0f;
    for (int k = 0; k < HID; ++k) s += msg_b[k] * upd_w[j * HID + k];
    Cbuf[9 + j] = s;
  }
}

// ---------------- MLP trunk: alpha -> x (16 x 256) ----------------
__device__ __forceinline__ void resblock(float* A, float* T,
                                         const float* __restrict__ w1,
                                         const float* __restrict__ b1,
                                         const float* __restrict__ w2,
                                         const float* __restrict__ b2,
                                         int j) {
  // T = relu(A @ w1^T + b1)
  {
    const float* wr = w1 + j * KDIM;
    float bb = b1[j];
    float acc[BATCH];
    for (int b = 0; b < BATCH; ++b) acc[b] = bb;
    for (int k = 0; k < KDIM; ++k) {
      float w = wr[k];
      for (int b = 0; b < BATCH; ++b) acc[b] += A[b * KDIM + k] * w;
    }
    for (int b = 0; b < BATCH; ++b) T[b * KDIM + j] = fmaxf(acc[b], 0.0f);
  }
  __syncthreads();
  // A = relu(T @ w2^T + b2 + A)   (thread j only touches column j of A)
  {
    const float* wr = w2 + j * KDIM;
    float bb = b2[j];
    float acc[BATCH];
    for (int b = 0; b < BATCH; ++b) acc[b] = bb + A[b * KDIM + j];
    for (int k = 0; k < KDIM; ++k) {
      float w = wr[k];
      for (int b = 0; b < BATCH; ++b) acc[b] += T[b * KDIM + k] * w;
    }
    for (int b = 0; b < BATCH; ++b) A[b * KDIM + j] = fmaxf(acc[b], 0.0f);
  }
  __syncthreads();
}

__global__ __launch_bounds__(256) void trunk_kernel(
    const float* __restrict__ alpha,
    const float* __restrict__ w_in, const float* __restrict__ b_in,
    const float* __restrict__ w1a, const float* __restrict__ b1a,
    const float* __restrict__ w2a, const float* __restrict__ b2a,
    const float* __restrict__ w1b, const float* __restrict__ b1b,
    const float* __restrict__ w2b, const float* __restrict__ b2b,
    float* __restrict__ x_out) {
  __shared__ float A[BATCH * KDIM];
  __shared__ float T[BATCH * KDIM];
  int j = threadIdx.x;  // 0..255
  float wj = w_in[j], bj = b_in[j];
  for (int b = 0; b < BATCH; ++b)
    A[b * KDIM + j] = fmaxf(alpha[b] * wj + bj, 0.0f);
  __syncthreads();
  resblock(A, T, w1a, b1a, w2a, b2a, j);
  resblock(A, T, w1b, b1b, w2b, b2b, j);
  for (int b = 0; b < BATCH; ++b) x_out[b * KDIM + j] = A[b * KDIM + j];
}

// ---------------- graph branch ----------------
__global__ void deg_kernel(const int* __restrict__ bonds, float* __restrict__ deg) {
  int e = blockIdx.x * blockDim.x + threadIdx.x;
  if (e >= NBONDS) return;
  int s = bonds[2 * e], d = bonds[2 * e + 1];
  atomicAdd(&deg[s], 1.0f);
  atomicAdd(&deg[d], 1.0f);
}

__global__ void scatter_kernel(const int* __restrict__ bonds,
                               const float* __restrict__ h,
                               float* __restrict__ sum) {
  int tid = blockIdx.x * blockDim.x + threadIdx.x;
  if (tid >= NBONDS * BATCH) return;
  int e = tid >> 4;
  int b = tid & 15;
  int s = bonds[2 * e], d = bonds[2 * e + 1];
  const float* hb = h + (size_t)b * COLS;
  float* sb = sum + (size_t)b * COLS;
  float hs0 = hb[s * 3 + 0], hs1 = hb[s * 3 + 1], hs2 = hb[s * 3 + 2];
  float hd0 = hb[d * 3 + 0], hd1 = hb[d * 3 + 1], hd2 = hb[d * 3 + 2];
  atomicAdd(&sb[d * 3 + 0], hs0);
  atomicAdd(&sb[d * 3 + 1], hs1);
  atomicAdd(&sb[d * 3 + 2], hs2);
  atomicAdd(&sb[s * 3 + 0], hd0);
  atomicAdd(&sb[s * 3 + 1], hd1);
  atomicAdd(&sb[s * 3 + 2], hd2);
}

__global__ void update_kernel(const float* __restrict__ h_src,
                              const float* __restrict__ sum,
                              const float* __restrict__ deg,
                              const float* __restrict__ Cbuf,
                              const float* __restrict__ upd_b,
                              float* __restrict__ h_dst) {
  int tid = blockIdx.x * blockDim.x + threadIdx.x;
  if (tid >= BATCH * NATOMS) return;
  int b = tid / NATOMS;
  int n = tid - b * NATOMS;
  size_t idx = (size_t)b * COLS + (size_t)n * 3;
  float dg = deg[n];
  float inv = 1.0f / fmaxf(dg, 1.0f);
  float m0 = sum[idx + 0] * inv;
  float m1 = sum[idx + 1] * inv;
  float m2 = sum[idx + 2] * inv;
  float has = (dg > 0.0f) ? 1.0f : 0.0f;
  // delta[j] = (sum_i m_i * C[i][j] + cvec[j]) masked; upd_b unconditional
  float d0 = (m0 * Cbuf[0] + m1 * Cbuf[3] + m2 * Cbuf[6] + Cbuf[9])  * has;
  float d1 = (m0 * Cbuf[1] + m1 * Cbuf[4] + m2 * Cbuf[7] + Cbuf[10]) * has;
  float d2 = (m0 * Cbuf[2] + m1 * Cbuf[5] + m2 * Cbuf[8] + Cbuf[11]) * has;
  h_dst[idx + 0] = h_src[idx + 0] + upd_b[0] + d0;
  h_dst[idx + 1] = h_src[idx + 1] + upd_b[1] + d1;
  h_dst[idx + 2] = h_src[idx + 2] + upd_b[2] + d2;
}

// ---------------- fused WMMA GEMM + epilogue ----------------
// out[b][col] = (x @ w_out^T)[b][col] + b_out[col] + (H @ go_w^T + go_b)[b][col]
// One 16x16 D-tile per wave: M = 16 batches, K = 256, 64 x V_WMMA_F32_16X16X4_F32.
__global__ __launch_bounds__(256) void gemm_kernel(
    const float* __restrict__ xw,
    const float* __restrict__ w_out, const float* __restrict__ b_out,
    const float* __restrict__ H,
    const float* __restrict__ go_w, const float* __restrict__ go_b,
    float* __restrict__ out) {
  __shared__ float sx[BATCH * KDIM];  // 16 KB activation stage
  for (int i = threadIdx.x; i < BATCH * KDIM; i += blockDim.x) sx[i] = xw[i];
  __syncthreads();

  int wave = threadIdx.x >> 5;     // 8 waves per block
  int lane = threadIdx.x & 31;
  int tile = blockIdx.x * 8 + wave;
  if (tile * 16 >= COLS) return;   // wave-uniform exit: EXEC all-ones for WMMA

  int n0 = lane & 15;
  int half = lane >> 4;            // 0: K pair {k0,k0+1}; 1: {k0+2,k0+3}
  int colbase = tile * 16;
  int col = colbase + n0;

  // B-matrix lane stream: w_out row `col`, contiguous float2 along K
  const float* wrow = w_out + (size_t)col * KDIM + 2 * half;
  // A-matrix lane stream: x row n0 (batch), contiguous float2 along K
  const float* xrow = sx + n0 * KDIM + 2 * half;

  v8f acc = {0.f, 0.f, 0.f, 0.f, 0.f, 0.f, 0.f, 0.f};
#pragma unroll 8
  for (int k0 = 0; k0 < KDIM; k0 += 4) {
    v2f a  = *(const v2f*)(xrow + k0);
    v2f bm = *(const v2f*)(wrow + k0);
    acc = __builtin_amdgcn_wmma_f32_16x16x4_f32(
        /*neg_a=*/false, a, /*neg_b=*/false, bm,
        /*c_mod=*/(short)0, acc, /*reuse_a=*/false, /*reuse_b=*/false);
  }

  // epilogue: D layout — VGPR v, lanes 0-15: (b=v, col), lanes 16-31: (b=v+8, col)
  float bo = b_out[col];
  int atom = col / 3;
  int c = col - atom * 3;
  float g0 = go_w[c * 3 + 0], g1 = go_w[c * 3 + 1], g2 = go_w[c * 3 + 2];
  float gb = go_b[c];
#pragma unroll
  for (int v = 0; v < 8; ++v) {
    int b = v + 8 * half;
    const float* hp = H + (size_t)b * COLS + (size_t)atom * 3;
    float g = gb + hp[0] * g0 + hp[1] * g1 + hp[2] * g2;
    out[(size_t)b * COLS + col] = acc[v] + bo + g;
  }
}

extern "C" void kernel_launch(void* const* d_in, const int* in_sizes, int n_in,
                              void* d_out, int out_size, void* d_ws, size_t ws_size,
                              hipStream_t stream) {
  const float* alpha  = (const float*)d_in[0];
  const float* base   = (const float*)d_in[1];
  const int*   bonds  = (const int*)  d_in[2];
  const float* w_in   = (const float*)d_in[3];
  const float* b_in   = (const float*)d_in[4];
  const float* rb1_w1 = (const float*)d_in[5];
  const float* rb1_b1 = (const float*)d_in[6];
  const float* rb1_w2 = (const float*)d_in[7];
  const float* rb1_b2 = (const float*)d_in[8];
  const float* rb2_w1 = (const float*)d_in[9];
  const float* rb2_b1 = (const float*)d_in[10];
  const float* rb2_w2 = (const float*)d_in[11];
  const float* rb2_b2 = (const float*)d_in[12];
  const float* w_out  = (const float*)d_in[13];
  const float* b_out  = (const float*)d_in[14];
  const float* msg_w  = (const float*)d_in[15];
  const float* msg_b  = (const float*)d_in[16];
  const float* upd_w  = (const float*)d_in[17];
  const float* upd_b  = (const float*)d_in[18];
  const float* go_w   = (const float*)d_in[19];
  const float* go_b   = (const float*)d_in[20];

  float* ws   = (float*)d_ws;
  float* xw   = ws + WS_X;
  float* Cbuf = ws + WS_C;
  float* deg  = ws + WS_DEG;
  float* sum  = ws + WS_SUM;
  float* H    = ws + WS_H;
  float* out  = (float*)d_out;

  // zero deg + sum (contiguous: WS_SUM == WS_DEG + NATOMS)
  zero_kernel<<<2048, 256, 0, stream>>>(deg, NATOMS + BATCH * COLS);
  fold_kernel<<<1, 64, 0, stream>>>(msg_w, msg_b, upd_w, Cbuf);
  trunk_kernel<<<1, 256, 0, stream>>>(alpha, w_in, b_in,
                                      rb1_w1, rb1_b1, rb1_w2, rb1_b2,
                                      rb2_w1, rb2_b1, rb2_w2, rb2_b2, xw);
  deg_kernel<<<(NBONDS + 255) / 256, 256, 0, stream>>>(bonds, deg);

  // iteration 1: baseline -> H
  scatter_kernel<<<(NBONDS * BATCH + 255) / 256, 256, 0, stream>>>(bonds, base, sum);
  update_kernel<<<(BATCH * NATOMS + 255) / 256, 256, 0, stream>>>(base, sum, deg, Cbuf, upd_b, H);

  // iteration 2: H -> H (re-zero sums first)
  zero_kernel<<<2048, 256, 0, stream>>>(sum, BATCH * COLS);
  scatter_kernel<<<(NBONDS * BATCH + 255) / 256, 256, 0, stream>>>(bonds, H, sum);
  update_kernel<<<(BATCH * NATOMS + 255) / 256, 256, 0, stream>>>(H, sum, deg, Cbuf, upd_b, H);

  // fused output GEMM (9375 tiles, 8 tiles/block)
  gemm_kernel<<<(COLS / 16 + 7) / 8, 256, 0, stream>>>(xw, w_out, b_out, H, go_w, go_b, out);
}